// CtdetLoss_41824391529152
// MI455X (gfx1250) — compile-verified
//
#include <hip/hip_runtime.h>
#include <hip/hip_bf16.h>

#define O_ 2
#define B_ 16
#define C_ 80
#define H_ 128
#define W_ 128
#define K_ 64
#define HW_ (H_*W_)
#define SPAD 132   // padded LDS row: bank = (4h + x) % 64 -> conflict-free B-frag loads

typedef __attribute__((ext_vector_type(2))) float v2f;
typedef __attribute__((ext_vector_type(8))) float v8f;
typedef __attribute__((ext_vector_type(4))) int   v4i;
typedef __attribute__((address_space(1))) v4i* gb128_p;   // global 128b
typedef __attribute__((address_space(3))) v4i* lb128_p;   // LDS 128b

#if defined(__has_builtin)
#if __has_builtin(__builtin_amdgcn_global_load_async_to_lds_b128)
#define HAVE_ASYNC_LDS 1
#endif
#if __has_builtin(__builtin_amdgcn_s_wait_asynccnt)
#define HAVE_WAIT_ASYNC 1
#endif
#endif

__device__ __forceinline__ void wait_asynccnt0() {
#if defined(HAVE_WAIT_ASYNC)
    __builtin_amdgcn_s_wait_asynccnt(0);
#else
    asm volatile("s_wait_asynccnt 0x0" ::: "memory");
#endif
}

// ---------------------------------------------------------------------------
// Kernel 1: S0[o,b,h,w] = sum_c log(1-p)*p*p   (p clipped to [1e-4, 1-1e-4])
// 168 MB coalesced reads -> HBM-bound (~7 us @ 23.3 TB/s). One log/elem.
// ---------------------------------------------------------------------------
__global__ void __launch_bounds__(256) s0_kernel(const float* __restrict__ out_hm,
                                                 float* __restrict__ s0) {
    int idx = blockIdx.x * 256 + threadIdx.x;       // over O*B*H*W
    if (idx >= O_ * B_ * HW_) return;
    int pix = idx & (HW_ - 1);
    int ob  = idx >> 14;                            // HW_ = 16384 = 2^14
    const float* base = out_hm + (size_t)ob * C_ * HW_ + pix;
    float acc = 0.f;
#pragma unroll 4
    for (int c = 0; c < C_; ++c) {
        float p = base[(size_t)c * HW_];
        p = fminf(fmaxf(p, 1e-4f), 1.f - 1e-4f);
        acc += __logf(1.f - p) * p * p;
    }
    s0[idx] = acc;
}

// ---------------------------------------------------------------------------
// Kernel 2: one workgroup per (o,b).
//   Phase A: async global->LDS staging of S0 (ASYNCcnt-tracked, no VGPR trip)
//   Phase B: R[k,h] = sum_x wx[k,x]*S0[h,x] via v_wmma_f32_16x16x4_f32,
//            then dot with wy[k,h] -> window sum of S0 per object.
//   Phase C: per-object class corrections + L1 wh/off gathers.
// ---------------------------------------------------------------------------
__global__ void __launch_bounds__(256) obj_kernel(
    const float* __restrict__ out_hm, const float* __restrict__ out_wh,
    const float* __restrict__ out_reg, const float* __restrict__ hm,
    const float* __restrict__ wh, const float* __restrict__ reg,
    const int* __restrict__ cxcy, const int* __restrict__ cls_idx,
    const int* __restrict__ ind, const float* __restrict__ s0g,
    float* __restrict__ tot_ws, float* __restrict__ hm_ws,
    float* __restrict__ wh_ws, float* __restrict__ off_ws)
{
    __shared__ float sS0[64 * SPAD];   // 64 rows per pass (2 passes over H)
    __shared__ float sWin[K_];
    __shared__ int sx0[K_], sx1[K_], sy0[K_], sy1[K_], scls[K_];

    const int ob = blockIdx.x;
    const int o  = ob / B_;
    const int b  = ob % B_;
    const int tid  = threadIdx.x;
    const int lane = tid & 31;
    const int wave = tid >> 5;
    const int lr   = lane & 15;

    // per-object spatial windows (mirrors the int arithmetic + clamps)
    if (tid < K_) {
        int cx = cxcy[(b * K_ + tid) * 2 + 0];
        int cy = cxcy[(b * K_ + tid) * 2 + 1];
        int wp = (int)(wh[(b * K_ + tid) * 2 + 0] * 0.5f);
        int hp = (int)(wh[(b * K_ + tid) * 2 + 1] * 0.5f);
        sy0[tid] = max(1, cy - hp / 2 - 1);
        sy1[tid] = min(H_ - 1, cy + hp / 2 + 1);
        sx0[tid] = max(1, cx - wp / 2 - 1);
        sx1[tid] = min(W_ - 1, cx + wp / 2 + 1);
        scls[tid] = cls_idx[b * K_ + tid];
        sWin[tid] = 0.f;
    }
    __syncthreads();

    // ---- Phase B: WMMA mask-matmul ----
    const int kt = wave >> 1;                 // A (object) tile: 0..3
    const int halfsel = (lane < 16) ? 0 : 2;  // A/B K-halves across lane halves
    const int am = kt * 16 + lr;              // A row = object index
    const int aX0 = sx0[am], aX1 = sx1[am];
    const int krowoff = (lane < 16) ? 0 : 8;  // D-fragment M split

    for (int ph = 0; ph < 2; ++ph) {
        // stage 64 rows of S0 into LDS (128b async transfers, 528B row pitch)
        const float4* src = (const float4*)(s0g + (size_t)ob * HW_ + ph * 64 * W_);
        for (int i = tid; i < 64 * W_ / 4; i += 256) {
            int h = (i * 4) >> 7, w = (i * 4) & (W_ - 1);
#if defined(HAVE_ASYNC_LDS)
            __builtin_amdgcn_global_load_async_to_lds_b128(
                (gb128_p)(src + i), (lb128_p)&sS0[h * SPAD + w], 0, 0);
#else
            *(float4*)&sS0[h * SPAD + w] = src[i];
#endif
        }
#if defined(HAVE_ASYNC_LDS)
        wait_asynccnt0();
#endif
        __syncthreads();

        const int ht0 = ph * 4 + (wave & 1) * 2;   // this wave's 2 h-tiles
        const float* rowp0 = &sS0[(((wave & 1) * 2 + 0) * 16 + lr) * SPAD];
        const float* rowp1 = &sS0[(((wave & 1) * 2 + 1) * 16 + lr) * SPAD];
        v8f acc0 = {}; v8f acc1 = {};

        for (int x4 = 0; x4 < W_; x4 += 4) {
            int kk = x4 + halfsel;
            v2f a;
            a.x = (kk     >= aX0 && kk     < aX1) ? 1.f : 0.f;
            a.y = (kk + 1 >= aX0 && kk + 1 < aX1) ? 1.f : 0.f;
            v2f b0 = *(const v2f*)&rowp0[kk];      // ds_load_b64, 8B aligned
            v2f b1 = *(const v2f*)&rowp1[kk];
            acc0 = __builtin_amdgcn_wmma_f32_16x16x4_f32(
                       false, a, false, b0, (short)0, acc0, false, false);
            acc1 = __builtin_amdgcn_wmma_f32_16x16x4_f32(
                       false, a, false, b1, (short)0, acc1, false, false);
        }

        // apply wy and reduce the 16 lanes sharing each object k
#pragma unroll
        for (int t = 0; t < 2; ++t) {
            const v8f acc = t ? acc1 : acc0;
            const int h = (ht0 + t) * 16 + lr;
#pragma unroll
            for (int v = 0; v < 8; ++v) {
                int k = kt * 16 + v + krowoff;
                float val = (h >= sy0[k] && h < sy1[k]) ? acc[v] : 0.f;
                val += __shfl_xor(val, 8, 32);
                val += __shfl_xor(val, 4, 32);
                val += __shfl_xor(val, 2, 32);
                val += __shfl_xor(val, 1, 32);
                if (lr == 0) atomicAdd(&sWin[k], val);   // ds_add_f32
            }
        }
        __syncthreads();
    }

    // ---- Phase C: per-object focal corrections + L1 gathers ----
    const float inv2 = 1.0f / (2.0f + 1e-4f);
    for (int k = wave; k < K_; k += 8) {
        const int x0 = sx0[k], x1 = sx1[k], y0 = sy0[k], y1 = sy1[k];
        const int cls = scls[k];
        const int wdt = x1 - x0;
        const int nw  = (y1 - y0) * wdt;
        const float* hmb = hm + (size_t)(b * C_ + cls) * HW_;
        const float* pb  = out_hm + (size_t)((o * B_ + b) * C_ + cls) * HW_;
        float pos = 0.f, corr = 0.f;
        int   npos = 0;
        for (int i = lane; i < nw; i += 32) {
            int y = y0 + i / wdt, x = x0 + i % wdt;
            int off = y * W_ + x;
            float g = hmb[off];
            float p = fminf(fmaxf(pb[off], 1e-4f), 1.f - 1e-4f);
            float omp  = 1.f - p;
            float neg0 = __logf(omp) * p * p;
            corr -= neg0;                       // remove class-channel S0 term
            if (g == 1.0f) {                    // focal positive
                pos += __logf(p) * omp * omp;
                npos += 1;
            } else {                            // focal negative vs real gt
                float q = 1.f - g, q2 = q * q;
                corr += neg0 * q2 * q2;
            }
        }
        for (int s = 16; s >= 1; s >>= 1) {
            pos  += __shfl_xor(pos,  s, 32);
            corr += __shfl_xor(corr, s, 32);
            npos += __shfl_xor(npos, s, 32);
        }
        if (lane == 0) {
            float neg_sum = sWin[k] + corr;
            float hm_l = (npos > 0) ? (-(pos + neg_sum) / (float)npos) : (-neg_sum);
            int id = ind[b * K_ + k];
            const float* owb = out_wh  + (size_t)(o * B_ + b) * 2 * HW_;
            const float* orb = out_reg + (size_t)(o * B_ + b) * 2 * HW_;
            float w0 = wh[(b * K_ + k) * 2 + 0], w1 = wh[(b * K_ + k) * 2 + 1];
            float r0 = reg[(b * K_ + k) * 2 + 0], r1 = reg[(b * K_ + k) * 2 + 1];
            float wh_l  = (fabsf(owb[id] - w0) + fabsf(owb[HW_ + id] - w1)) * inv2;
            float off_l = (fabsf(orb[id] - r0) + fabsf(orb[HW_ + id] - r1)) * inv2;
            float tot = hm_l + 0.1f * wh_l + off_l;     // HM_W=1, WH_W=0.1, OFF_W=1
            int oi = (o * B_ + b) * K_ + k;
            tot_ws[oi] = tot; hm_ws[oi] = hm_l; wh_ws[oi] = wh_l; off_ws[oi] = off_l;
        }
    }
}

// ---------------------------------------------------------------------------
// Kernel 3: FSAF argmin over O, masked sum, /B. 1024 threads = B*K items.
// ---------------------------------------------------------------------------
__global__ void __launch_bounds__(1024) reduce_kernel(
    const float* __restrict__ tot_ws, const float* __restrict__ hm_ws,
    const float* __restrict__ wh_ws, const float* __restrict__ off_ws,
    const int* __restrict__ reg_mask, float* __restrict__ out)
{
    __shared__ float sm[32 * 4];
    int t = threadIdx.x;
    int b = t >> 6, k = t & 63;
    int i0 = b * K_ + k;            // (o=0,b,k)
    int i1 = (B_ + b) * K_ + k;     // (o=1,b,k)
    float t0 = tot_ws[i0], t1 = tot_ws[i1];
    int ii = (t0 <= t1) ? i0 : i1;  // argmin picks first on tie
    float m  = (float)reg_mask[b * K_ + k];
    float vt = tot_ws[ii] * m, vh = hm_ws[ii] * m;
    float vw = wh_ws[ii] * m,  vo = off_ws[ii] * m;
    for (int s = 16; s >= 1; s >>= 1) {
        vt += __shfl_xor(vt, s, 32);
        vh += __shfl_xor(vh, s, 32);
        vw += __shfl_xor(vw, s, 32);
        vo += __shfl_xor(vo, s, 32);
    }
    int lane = t & 31, wave = t >> 5;
    if (lane == 0) {
        sm[wave * 4 + 0] = vt; sm[wave * 4 + 1] = vh;
        sm[wave * 4 + 2] = vw; sm[wave * 4 + 3] = vo;
    }
    __syncthreads();
    if (t < 4) {
        float s = 0.f;
        for (int w = 0; w < 32; ++w) s += sm[w * 4 + t];
        out[t] = s / (float)B_;     // order: loss, hm_loss, wh_loss, off_loss
    }
}

extern "C" void kernel_launch(void* const* d_in, const int* in_sizes, int n_in,
                              void* d_out, int out_size, void* d_ws, size_t ws_size,
                              hipStream_t stream) {
    const float* out_hm  = (const float*)d_in[0];
    const float* out_wh  = (const float*)d_in[1];
    const float* out_reg = (const float*)d_in[2];
    const float* hm      = (const float*)d_in[3];
    const float* wh      = (const float*)d_in[4];
    const float* reg     = (const float*)d_in[5];
    const int*   cxcy    = (const int*)d_in[6];
    const int*   cls_idx = (const int*)d_in[7];
    const int*   ind     = (const int*)d_in[8];
    const int*   reg_mask= (const int*)d_in[9];
    float* out = (float*)d_out;

    // workspace: S0 (2 MB) + 4 per-object result arrays (8K floats)
    float* s0     = (float*)d_ws;
    float* tot_ws = s0 + (size_t)O_ * B_ * HW_;
    float* hm_ws  = tot_ws + O_ * B_ * K_;
    float* wh_ws  = hm_ws  + O_ * B_ * K_;
    float* off_ws = wh_ws  + O_ * B_ * K_;

    int n1 = O_ * B_ * HW_;
    s0_kernel<<<(n1 + 255) / 256, 256, 0, stream>>>(out_hm, s0);
    obj_kernel<<<O_ * B_, 256, 0, stream>>>(out_hm, out_wh, out_reg, hm, wh, reg,
                                            cxcy, cls_idx, ind, s0,
                                            tot_ws, hm_ws, wh_ws, off_ws);
    reduce_kernel<<<1, 1024, 0, stream>>>(tot_ws, hm_ws, wh_ws, off_ws,
                                          reg_mask, out);
}